// ProceduralMemory_81535659147884
// MI455X (gfx1250) — compile-verified
//
#include <hip/hip_runtime.h>

typedef float v2f __attribute__((ext_vector_type(2)));
typedef float v8f __attribute__((ext_vector_type(8)));

namespace {
constexpr int kBS = 8;
constexpr int kN  = 4096;
constexpr int kB  = 8;
constexpr int kR  = 64;
constexpr int kD  = 256;

constexpr int kWaves     = 8;
constexpr int kThreads   = kWaves * 32;     // 256
constexpr int kRowsPerWG = kWaves * 16;     // 128
constexpr int kPitch     = 68;              // 64 + 4 pad -> pitch % 64 == 4 (conflict-free b64 frags)
constexpr int kChunk     = kR * kPitch;     // floats per staging buffer (17.4 KB)
}

// Low 32 bits of a generic pointer into LDS == wave-relative LDS byte offset
// (aperture high bits live in [63:32]; hardware maps LDS by discarding them).
__device__ __forceinline__ uint32_t lds_off(const void* p) {
    return (uint32_t)(uintptr_t)p;
}

__device__ __forceinline__ void async_wait_all() {
    asm volatile("s_wait_asynccnt 0x0" ::: "memory");
}

__global__ __launch_bounds__(kThreads)
void holo_read_wmma(const float* __restrict__ q,
                    const float* __restrict__ pmK,
                    const float* __restrict__ pmV,
                    const float* __restrict__ pma,
                    float* __restrict__ out)
{
    __shared__ float KV[2][kChunk];              // 2 x 17.4 KB double-buffered K / V^T chunks
    __shared__ float Ws[kWaves * 16 * kPitch];   // 34.8 KB per-wave weighted scores [16][64] padded
    __shared__ float invSh[kWaves * 16];
    __shared__ float aSh[kR];

    const int tid   = threadIdx.x;
    const int lane  = tid & 31;
    const int wave  = tid >> 5;
    const int l16   = lane & 15;
    const int khalf = (lane >> 4) * 2;           // lanes 16-31 hold K = base+2,+3
    const int mhi   = (lane >> 4) * 8;           // D-matrix: lanes 16-31 hold M = v+8

    const int sb    = blockIdx.x >> 5;           // (s,b) pair; consecutive blocks share it (L2 reuse)
    const int tileI = blockIdx.x & 31;
    const int s  = sb >> 3;
    const int b  = sb & 7;
    const int n0 = tileI * kRowsPerWG;

    const float* Kg = pmK + (size_t)sb * kR * kD;
    const float* Vg = pmV + (size_t)sb * kR * kD;

    const uint32_t kvBase0 = lds_off(&KV[0][0]);
    const uint32_t kvBase1 = lds_off(&KV[1][0]);

    if (tid < kR) aSh[tid] = pma[(size_t)sb * kR + tid];

    // --- async DMA staging: K chunk (row-major, padded pitch) ---
    auto stageK = [&](int dc, uint32_t base) {
        for (int i = tid; i < kR * 16; i += kThreads) {          // 4 iters/thread
            const int r  = i >> 4;
            const int c4 = i & 15;
            const float*   src = Kg + (size_t)r * kD + dc * 64 + c4 * 4;
            const uint32_t dst = base + (uint32_t)(r * kPitch + c4 * 4) * 4u;
            asm volatile("global_load_async_to_lds_b128 %0, %1, off"
                         :: "v"(dst), "v"(src) : "memory");
        }
    };
    // --- async DMA staging: V chunk transposed via per-lane scattered LDS dests ---
    auto stageVT = [&](int dc, uint32_t base) {
        for (int i = tid; i < kR * 16; i += kThreads) {
            const int r  = i >> 4;
            const int c4 = i & 15;
            const float* src = Vg + (size_t)r * kD + dc * 64 + c4 * 4;
            #pragma unroll
            for (int j = 0; j < 4; ++j) {
                const uint32_t dst = base + (uint32_t)((c4 * 4 + j) * kPitch + r) * 4u;
                asm volatile("global_load_async_to_lds_b32 %0, %1, off"
                             :: "v"(dst), "v"(src + j) : "memory");
            }
        }
    };

    const int wrow0 = n0 + wave * 16;
    const float* qrow = q + (((size_t)s * kN + (size_t)(wrow0 + l16)) * kB + b) * kD;

    v8f acc0 = {}, acc1 = {}, acc2 = {}, acc3 = {};
    float ssq = 0.f;

    // ---------------- Phase 1: S = q . K^T, D chunked by 64, double-buffered ----------------
    stageK(0, kvBase0);
    async_wait_all();
    __syncthreads();

    for (int dc = 0; dc < 4; ++dc) {
        const uint32_t curBase = (dc & 1) ? kvBase1 : kvBase0;
        const float*   KVc     = (dc & 1) ? &KV[1][0] : &KV[0][0];
        if (dc < 3) stageK(dc + 1, (dc & 1) ? kvBase0 : kvBase1);   // overlap DMA with WMMA
        (void)curBase;
        #pragma unroll 4
        for (int kk = 0; kk < 16; ++kk) {
            const int k = kk * 4 + khalf;
            v2f a = *(const v2f*)(qrow + dc * 64 + k);
            ssq += a.x * a.x + a.y * a.y;                 // free row-norm accumulation
            v2f b0 = *(const v2f*)(&KVc[(0 * 16 + l16) * kPitch + k]);
            v2f b1 = *(const v2f*)(&KVc[(1 * 16 + l16) * kPitch + k]);
            v2f b2 = *(const v2f*)(&KVc[(2 * 16 + l16) * kPitch + k]);
            v2f b3 = *(const v2f*)(&KVc[(3 * 16 + l16) * kPitch + k]);
            acc0 = __builtin_amdgcn_wmma_f32_16x16x4_f32(false, a, false, b0, (short)0, acc0, false, false);
            acc1 = __builtin_amdgcn_wmma_f32_16x16x4_f32(false, a, false, b1, (short)0, acc1, false, false);
            acc2 = __builtin_amdgcn_wmma_f32_16x16x4_f32(false, a, false, b2, (short)0, acc2, false, false);
            acc3 = __builtin_amdgcn_wmma_f32_16x16x4_f32(false, a, false, b3, (short)0, acc3, false, false);
        }
        async_wait_all();      // next chunk landed
        __syncthreads();       // everyone done reading current chunk
    }

    // Start streaming the first V^T chunk while we fold norms / build W.
    stageVT(0, kvBase0);       // KV[0] last read at dc=2; all readers passed the dc=2 barrier

    // lanes L and L^16 jointly covered row l16 of the strip
    ssq += __shfl_xor(ssq, 16, 32);
    const float inv = 1.0f / fmaxf(sqrtf(ssq), 1e-8f);
    if (lane < 16) invSh[wave * 16 + lane] = inv;        // wave-private region; in-wave DS ordering

    // W = S * (1/||q||) * a  -> LDS in [M][r] layout (A-fragment shape for GEMM2)
    {
        float* wsW = &Ws[wave * 16 * kPitch];
        #pragma unroll
        for (int t = 0; t < 4; ++t) {
            const v8f accT = (t == 0) ? acc0 : (t == 1) ? acc1 : (t == 2) ? acc2 : acc3;
            const float aw = aSh[t * 16 + l16];
            #pragma unroll
            for (int v = 0; v < 8; ++v) {
                const int M = v + mhi;
                wsW[M * kPitch + t * 16 + l16] = accT[v] * invSh[wave * 16 + M] * aw;
            }
        }
    }

    async_wait_all();
    __syncthreads();           // V^T chunk 0 ready; Ws in place

    // ---------------- Phase 2: out = q * (W . V), output D chunked by 64 ----------------
    const float* wsA = &Ws[(wave * 16 + l16) * kPitch];
    const size_t rowStride = (size_t)kB * kD;
    const size_t outBase = (((size_t)s * kN + wrow0) * kB + b) * kD;

    for (int dc = 0; dc < 4; ++dc) {
        const float* KVc = (dc & 1) ? &KV[1][0] : &KV[0][0];
        if (dc < 3) stageVT(dc + 1, (dc & 1) ? kvBase0 : kvBase1);
        #pragma unroll 1
        for (int ntl = 0; ntl < 4; ++ntl) {
            v8f c = {};
            #pragma unroll 4
            for (int kk = 0; kk < 16; ++kk) {
                const int k = kk * 4 + khalf;             // k indexes r (contraction dim)
                v2f a  = *(const v2f*)(wsA + k);
                v2f bb = *(const v2f*)(&KVc[(ntl * 16 + l16) * kPitch + k]);
                c = __builtin_amdgcn_wmma_f32_16x16x4_f32(false, a, false, bb, (short)0, c, false, false);
            }
            const int d = dc * 64 + ntl * 16 + l16;
            #pragma unroll
            for (int v = 0; v < 8; ++v) {
                const size_t idx = outBase + (size_t)(v + mhi) * rowStride + (size_t)d;
                out[idx] = q[idx] * c[v];
            }
        }
        async_wait_all();
        __syncthreads();
    }
}

extern "C" void kernel_launch(void* const* d_in, const int* in_sizes, int n_in,
                              void* d_out, int out_size, void* d_ws, size_t ws_size,
                              hipStream_t stream) {
    const float* q   = (const float*)d_in[0];
    const float* pmK = (const float*)d_in[1];
    const float* pmV = (const float*)d_in[2];
    const float* pma = (const float*)d_in[3];
    float* out = (float*)d_out;

    const int nBlocks = (kBS * kB) * (kN / kRowsPerWG);  // 64 * 32 = 2048
    holo_read_wmma<<<nBlocks, kThreads, 0, stream>>>(q, pmK, pmV, pma, out);
}